// FeatureEnhancer_85847806313180
// MI455X (gfx1250) — compile-verified
//
#include <hip/hip_runtime.h>
#include <math.h>

typedef float v2f __attribute__((ext_vector_type(2)));
typedef float v8f __attribute__((ext_vector_type(8)));

#define WMMA_F32(a, b, c) \
  __builtin_amdgcn_wmma_f32_16x16x4_f32(false, (a), false, (b), (short)0, (c), false, false)

// ---------------------------------------------------------------------------
// K0a: fold wa1 into wq/wk:  WA = wa1[:, :C] @ wq,  WK = wa1[:, C:] @ wk
//      bA = wa1[:, :C] @ bq + ba1,  bK = wa1[:, C:] @ bk
// ---------------------------------------------------------------------------
__global__ void k_fuse_weights(const float* __restrict__ wa1, const float* __restrict__ wq,
                               const float* __restrict__ wk, const float* __restrict__ bq,
                               const float* __restrict__ bk, const float* __restrict__ ba1,
                               float* __restrict__ WA, float* __restrict__ WK,
                               float* __restrict__ bA, float* __restrict__ bK)
{
  int o = blockIdx.x, c = threadIdx.x;
  float sa = 0.f, sk = 0.f;
  for (int j = 0; j < 128; ++j) {
    sa += wa1[o * 256 + j]       * wq[j * 128 + c];
    sk += wa1[o * 256 + 128 + j] * wk[j * 128 + c];
  }
  WA[o * 128 + c] = sa;
  WK[o * 128 + c] = sk;
  if (c == 0) {
    float a = 0.f, b = 0.f;
    for (int j = 0; j < 128; ++j) {
      a += wa1[o * 256 + j]       * bq[j];
      b += wa1[o * 256 + 128 + j] * bk[j];
    }
    bA[o] = a + ba1[o];
    bK[o] = b;
  }
}

// ---------------------------------------------------------------------------
// K0b: conv weight transform  wo[o][c][ky][kx] -> wT[o][k3][c]  (contiguous K)
// ---------------------------------------------------------------------------
__global__ void k_prep_wT(const float* __restrict__ wo, float* __restrict__ wT)
{
  int idx = blockIdx.x * 256 + threadIdx.x;
  if (idx < 147456) {
    int c  = idx & 127;
    int k3 = (idx >> 7) % 9;
    int o  = idx / 1152;
    wT[idx] = wo[(o * 128 + c) * 9 + k3];
  }
}

// ---------------------------------------------------------------------------
// K1: (B,T,C,H,W) -> (b, y, x, t, c)   LDS-tiled (c,x) transpose per (b,t,y)
// ---------------------------------------------------------------------------
__global__ void k_transpose(const float* __restrict__ dec, const float* __restrict__ enc,
                            float* __restrict__ decT, float* __restrict__ encT)
{
  __shared__ float tile[32 * 129];
  const float* src = blockIdx.y ? enc : dec;
  float* dst       = blockIdx.y ? encT : decT;
  int gid = blockIdx.x;
  int b = gid >> 8, rem = gid & 255, t = rem >> 5, y = rem & 31;
  int tid = threadIdx.x, tx = tid & 31, ty = tid >> 5;
  size_t sbase = ((size_t)((b * 8 + t) * 128)) * 1024 + y * 32;
  for (int i = 0; i < 16; ++i) {
    int c = ty + i * 8;
    tile[tx * 129 + c] = src[sbase + (size_t)c * 1024 + tx];   // coalesced over x
  }
  __syncthreads();
  size_t dbase = ((size_t)(b * 1024 + y * 32)) * 1024 + t * 128;
  for (int i = 0; i < 4; ++i) {
    int xx = ty + i * 8;
    for (int j = 0; j < 4; ++j) {
      int c = tx + j * 32;
      dst[dbase + (size_t)xx * 1024 + c] = tile[xx * 129 + c]; // coalesced over c
    }
  }
}

// ---------------------------------------------------------------------------
// 16x128 @ 128x128^T GEMM on V_WMMA_F32_16X16X4_F32.
// Ash: LDS 16x128 (row m = p*8+t, channel-contiguous). W row-major [o][c].
// ---------------------------------------------------------------------------
__device__ __forceinline__ void gemm16x128(const float* Ash, const float* __restrict__ W,
                                           const float* __restrict__ bias, float* Osh)
{
  int lane = threadIdx.x & 31;
  int m    = lane & 15;
  int koff = (lane >> 4) * 2;   // lanes 0-15 carry K%4 in {0,1}; lanes 16-31 in {2,3}
  v2f a[32];
#pragma unroll
  for (int k = 0; k < 32; ++k)
    a[k] = *(const v2f*)(Ash + m * 128 + k * 4 + koff);
  for (int n = 0; n < 8; ++n) {
    int col = n * 16 + m;
    float bv = bias[col];
    v8f c;
#pragma unroll
    for (int j = 0; j < 8; ++j) c[j] = bv;
    const float* wrow = W + col * 128 + koff;
#pragma unroll
    for (int k = 0; k < 32; ++k) {
      v2f b = *(const v2f*)(wrow + k * 4);
      c = WMMA_F32(a[k], b, c);
    }
    int rb = (lane >> 4) * 8;
#pragma unroll
    for (int j = 0; j < 8; ++j) Osh[(rb + j) * 128 + col] = c[j];
  }
}

// ---------------------------------------------------------------------------
// K2: fused q/aq/ak/v GEMMs + additive attention + attn_sum, per pixel pair.
// One wave per 2 pixels (16 rows = 2 px * 8 timesteps).
// ---------------------------------------------------------------------------
__global__ void __launch_bounds__(32)
k_attn(const float* __restrict__ decT, const float* __restrict__ encT,
       const float* __restrict__ wq, const float* __restrict__ bq,
       const float* __restrict__ wv, const float* __restrict__ bv,
       const float* __restrict__ WA, const float* __restrict__ bA,
       const float* __restrict__ WK, const float* __restrict__ bK,
       const float* __restrict__ wa2, const float* __restrict__ ba2,
       float* __restrict__ hpre)
{
  __shared__ float sIn[2048], sQ[2048], sAQ[2048], sAK[2048], sV[2048];
  __shared__ float sW2[128], sAW[128];
  int lane = threadIdx.x;
  int pair = blockIdx.x;
  const float* dsrc = decT + (size_t)pair * 2048;
  const float* esrc = encT + (size_t)pair * 2048;

  *(float4*)(&sW2[lane * 4]) = *(const float4*)(wa2 + lane * 4);
  for (int i = 0; i < 16; ++i)
    *(float4*)(&sIn[i * 128 + lane * 4]) = *(const float4*)(dsrc + i * 128 + lane * 4);

  gemm16x128(sIn, wq, bq, sQ);    // q  = dec*wq^T + bq
  gemm16x128(sIn, WA, bA, sAQ);   // aq = dec*WA^T + (wa1q@bq + ba1)

  for (int i = 0; i < 16; ++i)
    *(float4*)(&sIn[i * 128 + lane * 4]) = *(const float4*)(esrc + i * 128 + lane * 4);

  gemm16x128(sIn, WK, bK, sAK);   // ak = enc*WK^T + wa1k@bk
  gemm16x128(sIn, wv, bv, sV);    // v  = enc*wv^T + bv

  // logits[d][e] = sum_c relu(aq+ak)*wa2 + ba2 ; w = sigmoid
  float ba2v = ba2[0];
  int p = lane >> 4;
  for (int j = 0; j < 4; ++j) {
    int pr = (lane & 15) * 4 + j;
    int d = pr >> 3, e = pr & 7;
    const float* paq = sAQ + (p * 8 + d) * 128;
    const float* pak = sAK + (p * 8 + e) * 128;
    float s = ba2v;
    for (int c = 0; c < 128; ++c)
      s += fmaxf(paq[c] + pak[c], 0.f) * sW2[c];
    sAW[p * 64 + d * 8 + e] = 1.f / (1.f + __expf(-s));
  }

  // hpre = q + (1/Te) * sum_e w[d,e]*v[e]   -> layout (b,d, y,x, c)
  size_t pix0 = (size_t)pair * 2;
  int b = (int)(pix0 >> 10), rem = (int)(pix0 & 1023);
  for (int r = 0; r < 16; ++r) {
    int pp = r >> 3, d = r & 7;
    for (int cc = 0; cc < 4; ++cc) {
      int c = lane + cc * 32;
      float acc = 0.f;
      for (int e = 0; e < 8; ++e)
        acc += sAW[pp * 64 + d * 8 + e] * sV[(pp * 8 + e) * 128 + c];
      hpre[(((size_t)(b * 8 + d)) * 1024 + rem + pp) * 128 + c] = sQ[r * 128 + c] + acc * 0.125f;
    }
  }
}

// ---------------------------------------------------------------------------
// K3/K5: group-norm stats per (b,t,g) -> per-channel scale/shift
// ---------------------------------------------------------------------------
__global__ void k_stats(const float* __restrict__ X, const float* __restrict__ g,
                        const float* __restrict__ be, float* __restrict__ scale,
                        float* __restrict__ shift)
{
  __shared__ float sS[256], sP[256];
  int gid = blockIdx.x, img = gid >> 3, grp = gid & 7;
  int tid = threadIdx.x;
  float s = 0.f, s2 = 0.f;
  const float* base = X + (size_t)img * 131072 + grp * 16;
  for (int f = tid; f < 16384; f += 256) {
    float v = base[(f >> 4) * 128 + (f & 15)];
    s += v; s2 += v * v;
  }
  sS[tid] = s; sP[tid] = s2;
  __syncthreads();
  for (int off = 128; off > 0; off >>= 1) {
    if (tid < off) { sS[tid] += sS[tid + off]; sP[tid] += sP[tid + off]; }
    __syncthreads();
  }
  if (tid < 16) {
    float mean = sS[0] * (1.f / 16384.f);
    float var  = sP[0] * (1.f / 16384.f) - mean * mean;
    float rstd = rsqrtf(var + 1e-5f);
    int c = grp * 16 + tid;
    float sc = rstd * g[c];
    scale[img * 128 + c] = sc;
    shift[img * 128 + c] = be[c] - mean * sc;
  }
}

// ---------------------------------------------------------------------------
// K4: implicit-GEMM 3x3 conv on WMMA.  One wave per 16-pixel row segment.
// tout = h + conv3x3(h) + bo   with h = hpre*scale1 + shift1
// ---------------------------------------------------------------------------
__global__ void __launch_bounds__(32)
k_conv(const float* __restrict__ hpre, const float* __restrict__ wT,
       const float* __restrict__ bo, const float* __restrict__ scale1,
       const float* __restrict__ shift1, float* __restrict__ tout)
{
  __shared__ float patch[3 * 18 * 128];
  int lane = threadIdx.x;
  int bx = blockIdx.x;
  int img = bx >> 6, rem = bx & 63, yy = rem >> 1, x0 = (rem & 1) * 16;
  float4 sc = *(const float4*)(scale1 + img * 128 + lane * 4);
  float4 sh = *(const float4*)(shift1 + img * 128 + lane * 4);
  const float* hb = hpre + (size_t)img * 131072;

  for (int ry = 0; ry < 3; ++ry) {
    int gy = yy + ry - 1;
    for (int col = 0; col < 18; ++col) {
      int gx = x0 + col - 1;
      float4 v = make_float4(0.f, 0.f, 0.f, 0.f);
      if (gy >= 0 && gy < 32 && gx >= 0 && gx < 32) {
        float4 r = *(const float4*)(hb + ((size_t)(gy * 32 + gx)) * 128 + lane * 4);
        v.x = r.x * sc.x + sh.x; v.y = r.y * sc.y + sh.y;
        v.z = r.z * sc.z + sh.z; v.w = r.w * sc.w + sh.w;
      }
      *(float4*)(&patch[(ry * 18 + col) * 128 + lane * 4]) = v;
    }
  }

  int m = lane & 15, koff = (lane >> 4) * 2, rb = (lane >> 4) * 8;
  v8f acc[8];
#pragma unroll
  for (int n = 0; n < 8; ++n) {
    float bv = bo[n * 16 + m];
#pragma unroll
    for (int j = 0; j < 8; ++j) acc[n][j] = bv;
  }

  for (int k3 = 0; k3 < 9; ++k3) {
    int ky = k3 / 3, kx = k3 - ky * 3;
    v2f a[32];
#pragma unroll
    for (int k = 0; k < 32; ++k)
      a[k] = *(const v2f*)(&patch[(ky * 18 + m + kx) * 128 + k * 4 + koff]);
#pragma unroll
    for (int n = 0; n < 8; ++n) {
      const float* wrow = wT + ((n * 16 + m) * 9 + k3) * 128 + koff;
#pragma unroll
      for (int k = 0; k < 32; ++k) {
        v2f b = *(const v2f*)(wrow + k * 4);
        acc[n] = WMMA_F32(a[k], b, acc[n]);
      }
    }
  }

  float* ob = tout + (size_t)img * 131072 + (size_t)(yy * 32 + x0) * 128;
#pragma unroll
  for (int n = 0; n < 8; ++n) {
    int c = n * 16 + m;
#pragma unroll
    for (int j = 0; j < 8; ++j) {
      int mm = rb + j;
      ob[mm * 128 + c] = acc[n][j] + patch[(18 + mm + 1) * 128 + c];
    }
  }
}

// ---------------------------------------------------------------------------
// K6: out(B,T,C,H,W) = t(b,t,y,x,c)*scale2 + shift2   (layout restore)
// ---------------------------------------------------------------------------
__global__ void k_final(const float* __restrict__ t, const float* __restrict__ scale2,
                        const float* __restrict__ shift2, float* __restrict__ out)
{
  int gidx = blockIdx.x * 256 + threadIdx.x;
  int pix = gidx & 1023;
  int c   = (gidx >> 10) & 127;
  int img = gidx >> 17;
  out[gidx] = t[((size_t)img * 1024 + pix) * 128 + c] * scale2[img * 128 + c] + shift2[img * 128 + c];
}

// ---------------------------------------------------------------------------
extern "C" void kernel_launch(void* const* d_in, const int* in_sizes, int n_in,
                              void* d_out, int out_size, void* d_ws, size_t ws_size,
                              hipStream_t stream)
{
  (void)in_sizes; (void)n_in; (void)out_size; (void)ws_size;
  const float* enc = (const float*)d_in[0];
  const float* dec = (const float*)d_in[1];
  const float* wq  = (const float*)d_in[2];
  const float* bq  = (const float*)d_in[3];
  const float* wk  = (const float*)d_in[4];
  const float* bk  = (const float*)d_in[5];
  const float* wv  = (const float*)d_in[6];
  const float* bv  = (const float*)d_in[7];
  const float* wa1 = (const float*)d_in[8];
  const float* ba1 = (const float*)d_in[9];
  const float* wa2 = (const float*)d_in[10];
  const float* ba2 = (const float*)d_in[11];
  const float* g1  = (const float*)d_in[12];
  const float* be1 = (const float*)d_in[13];
  const float* g2  = (const float*)d_in[14];
  const float* be2 = (const float*)d_in[15];
  const float* wo  = (const float*)d_in[16];
  const float* bo  = (const float*)d_in[17];
  float* out = (float*)d_out;
  float* ws  = (float*)d_ws;

  float* WA = ws;                 float* WK = ws + 16384;
  float* bA = ws + 32768;         float* bK = ws + 32896;
  float* wT = ws + 33024;                                  // 147456
  float* scale1 = ws + 180480;    float* shift1 = ws + 188672;
  float* scale2 = ws + 196864;    float* shift2 = ws + 205056;
  float* decT = ws + 262144;                               // 8388608 (reused as tout)
  float* encT = ws + 262144 + 8388608;                     // 8388608
  float* hpre = ws + 262144 + 2 * 8388608;                 // 8388608
  float* tout = decT;                                      // decT dead after k_attn

  k_fuse_weights<<<128, 128, 0, stream>>>(wa1, wq, wk, bq, bk, ba1, WA, WK, bA, bK);
  k_prep_wT<<<576, 256, 0, stream>>>(wo, wT);
  k_transpose<<<dim3(2048, 2), 256, 0, stream>>>(dec, enc, decT, encT);
  k_attn<<<4096, 32, 0, stream>>>(decT, encT, wq, bq, wv, bv, WA, bA, WK, bK, wa2, ba2, hpre);
  k_stats<<<512, 256, 0, stream>>>(hpre, g1, be1, scale1, shift1);
  k_conv<<<4096, 32, 0, stream>>>(hpre, wT, bo, scale1, shift1, tout);
  k_stats<<<512, 256, 0, stream>>>(tout, g2, be2, scale2, shift2);
  k_final<<<32768, 256, 0, stream>>>(tout, scale2, shift2, out);
}